// ConformerModel_47897475285549
// MI455X (gfx1250) — compile-verified
//
#include <hip/hip_runtime.h>

// Problem constants (from reference): B=4, T=512, U=128, D=512, V=1024
#define BB 4
#define TT 512
#define UU 128
#define DD 512
#define VV 1024
#define WSTRIDE (2 * DD)       // fc_weight row stride = 2D = 1024
#define M_ENC (BB * TT)        // 2048 encoder rows
#define M_DEC (BB * UU)        // 512 decoder rows

typedef float v2f __attribute__((ext_vector_type(2)));
typedef float v4f __attribute__((ext_vector_type(4)));
typedef float v8f __attribute__((ext_vector_type(8)));

// ---------------------------------------------------------------------------
// Kernel 1: fused GEMM via V_WMMA_F32_16X16X4_F32.
//   rows 0..2047   : enc_proj[m,v] = sum_d enc[m,d]   * W[v, d]
//   rows 2048..2559: dec_proj[m,v] = sum_d dec[m,d]   * W[v, D+d]
// One wave computes one 16x16 tile; K loop steps by 4 (128 WMMAs per tile).
//
// f32 WMMA fragment layouts (ISA 7.12.2):
//   A 16x4 : lane L(<16) M=L holds {K=0,K=1}; lane L(>=16) M=L-16 holds {K=2,K=3}
//            -> per-lane contiguous float2 at A[row] + k + 2*(lane>>4)
//   B 4x16 : lane L holds column N=L%16; VGPR0/1 = {K=0,K=1} (lanes 0-15),
//            {K=2,K=3} (lanes 16-31) -> per-lane contiguous float2 from W row N
//   C/D    : VGPR r, lane L -> M = r + 8*(L>>4), N = L%16
// ---------------------------------------------------------------------------
__global__ void __launch_bounds__(256)
joint_proj_wmma(const float* __restrict__ enc,   // (B*T, D)
                const float* __restrict__ dec,   // (B*U, D)
                const float* __restrict__ W,     // (V, 2D) row-major
                float* __restrict__ enc_proj,    // (B*T, V)
                float* __restrict__ dec_proj)    // (B*U, V)
{
    const int wid    = (blockIdx.x * blockDim.x + threadIdx.x) >> 5;
    const int lane   = threadIdx.x & 31;
    const int tile_m = wid >> 6;            // 0..159  (160 * 16 = 2560 rows)
    const int tile_n = wid & 63;            // 0..63   (64 * 16 = 1024 cols)
    const int mbase  = tile_m * 16;

    const float* Abase;
    float*       Obase;
    int          wOff;
    if (mbase < M_ENC) {
        Abase = enc + (size_t)mbase * DD;
        Obase = enc_proj + (size_t)mbase * VV;
        wOff  = 0;
    } else {
        Abase = dec + (size_t)(mbase - M_ENC) * DD;
        Obase = dec_proj + (size_t)(mbase - M_ENC) * VV;
        wOff  = DD;
    }

    const int khalf = (lane >> 4) * 2;                 // 0 for lanes 0-15, 2 for 16-31
    const float* arow = Abase + (size_t)(lane & 15) * DD + khalf;
    const float* brow = W + (size_t)(tile_n * 16 + (lane & 15)) * WSTRIDE + wOff + khalf;

    v8f acc = {};
#pragma unroll 8
    for (int k = 0; k < DD; k += 4) {
        v2f a = *(const v2f*)(arow + k);
        v2f b = *(const v2f*)(brow + k);
        // 8 args: (neg_a, A, neg_b, B, c_mod, C, reuse_a, reuse_b)
        acc = __builtin_amdgcn_wmma_f32_16x16x4_f32(
            false, a, false, b, (short)0, acc, false, false);
    }

    const int srow = (lane >> 4) * 8;
    const int scol = tile_n * 16 + (lane & 15);
#pragma unroll
    for (int r = 0; r < 8; ++r) {
        Obase[(size_t)(srow + r) * VV + scol] = acc[r];
    }
}

// ---------------------------------------------------------------------------
// Kernel 2: out[b,t,u,v] = enc_proj[b,t,v] + dec_proj[b,u,v]
// One block per (b,t); thread owns one float4 of V (256 threads = 1024 floats).
// enc value register-resident; dec read from L2 (2 MiB, resident); the 1 GiB
// output is written with non-temporal stores to keep it from thrashing L2.
// ---------------------------------------------------------------------------
__global__ void __launch_bounds__(256)
bcast_add(const float* __restrict__ enc_proj,
          const float* __restrict__ dec_proj,
          float* __restrict__ out)
{
    const int bt = blockIdx.x;              // 0 .. B*T-1
    const int b  = bt / TT;
    const int v4 = threadIdx.x;             // 0 .. 255

    const v4f e = ((const v4f*)(enc_proj + (size_t)bt * VV))[v4];
    const v4f* __restrict__ dptr = (const v4f*)(dec_proj + (size_t)b * UU * VV);
    v4f* __restrict__ optr = (v4f*)(out + (size_t)bt * UU * VV);

    const int VQ = VV / 4;
#pragma unroll 4
    for (int u = 0; u < UU; ++u) {
        v4f dv = dptr[u * VQ + v4];
        v4f o  = e + dv;
        __builtin_nontemporal_store(o, &optr[u * VQ + v4]);
    }
}

extern "C" void kernel_launch(void* const* d_in, const int* in_sizes, int n_in,
                              void* d_out, int out_size, void* d_ws, size_t ws_size,
                              hipStream_t stream) {
    const float* enc = (const float*)d_in[0];   // (4,512,512)
    const float* dec = (const float*)d_in[1];   // (4,128,512)
    const float* W   = (const float*)d_in[2];   // (1024,1024)
    float* out = (float*)d_out;                 // (4,512,128,1024)

    float* enc_proj = (float*)d_ws;                          // 8 MiB
    float* dec_proj = enc_proj + (size_t)M_ENC * VV;         // +2 MiB

    // GEMM: 160 row-tiles * 64 col-tiles = 10240 waves; 8 waves/block -> 1280 blocks
    joint_proj_wmma<<<1280, 256, 0, stream>>>(enc, dec, W, enc_proj, dec_proj);

    // Broadcast add: one block per (b,t)
    bcast_add<<<BB * TT, 256, 0, stream>>>(enc_proj, dec_proj, out);
}